// GCN_57045755625627
// MI455X (gfx1250) — compile-verified
//
#include <hip/hip_runtime.h>
#include <math.h>

#define NN 100000
#define NE 3200000
#define FIN 256
#define HID 16
#define NCLS 40

typedef __attribute__((ext_vector_type(2))) float v2f;
typedef __attribute__((ext_vector_type(8))) float v8f;

// ---------------- degree / normalization ----------------
__global__ __launch_bounds__(256) void deg_init_kernel(float* __restrict__ deg) {
    int i = blockIdx.x * 256 + threadIdx.x;
    if (i < NN) deg[i] = 1.0f;  // self-loop contributes 1 to in-degree
}

__global__ __launch_bounds__(256) void deg_count_kernel(const int* __restrict__ col,
                                                        float* __restrict__ deg) {
    int e = blockIdx.x * 256 + threadIdx.x;
    if (e < NE) atomicAdd(&deg[col[e]], 1.0f);
}

__global__ __launch_bounds__(256) void dinv_kernel(const float* __restrict__ deg,
                                                   float* __restrict__ dinv) {
    int i = blockIdx.x * 256 + threadIdx.x;
    if (i < NN) dinv[i] = rsqrtf(deg[i]);  // deg >= 1 always (self-loop)
}

// ---------------- GEMM1: xw1 = x @ W1 (fp32 WMMA 16x16x4) ----------------
// One wave handles TWO 16-row tiles (32 rows): one shared B fragment per k-step,
// two independent accumulator chains so back-to-back v_wmma need no delay slots.
// A: lane l holds x[row][k + 2*(l>>4) .. +1];  B: lane l holds W1[kk][l&15];
// C/D: VGPR j, lane l -> row (j + 8*(l>>4)), col (l&15).
__global__ __launch_bounds__(32) void gemm1_wmma_kernel(const float* __restrict__ x,
                                                        const float* __restrict__ W1,
                                                        const float* __restrict__ dinv,
                                                        float* __restrict__ xw1,
                                                        float* __restrict__ h1acc) {
    const int lane = threadIdx.x;
    const int m  = lane & 15;
    const int hi = lane >> 4;
    const int base = blockIdx.x * 32;
    const float* xr0 = x + (size_t)(base + m) * FIN;
    const float* xr1 = xr0 + (size_t)16 * FIN;
    v8f c0 = {0.f, 0.f, 0.f, 0.f, 0.f, 0.f, 0.f, 0.f};
    v8f c1 = {0.f, 0.f, 0.f, 0.f, 0.f, 0.f, 0.f, 0.f};
    for (int k = 0; k < FIN; k += 4) {
        const int kk = k + 2 * hi;
        v2f a0 = *(const v2f*)(xr0 + kk);   // contiguous 8B load
        v2f a1 = *(const v2f*)(xr1 + kk);
        v2f b;
        b.x = W1[(size_t)kk * HID + m];     // 16-lane coalesced 64B row
        b.y = W1[(size_t)(kk + 1) * HID + m];
        c0 = __builtin_amdgcn_wmma_f32_16x16x4_f32(false, a0, false, b,
                                                   (short)0, c0, false, false);
        c1 = __builtin_amdgcn_wmma_f32_16x16x4_f32(false, a1, false, b,
                                                   (short)0, c1, false, false);
    }
#pragma unroll
    for (int t = 0; t < 2; ++t) {
        const v8f& c = t ? c1 : c0;
#pragma unroll
        for (int j = 0; j < 8; ++j) {
            const int r = base + t * 16 + j + 8 * hi;
            const float v  = c[j];
            const float di = dinv[r];
            xw1[(size_t)r * HID + m]   = v;            // raw, for edge gather
            h1acc[(size_t)r * HID + m] = di * di * v;  // self-loop term of layer-1 agg
        }
    }
}

// ---------------- layer-1 edge scatter: h1acc[col] += dinv[row]*dinv[col]*xw1[row] ----
__global__ __launch_bounds__(256) void scatter1_kernel(const int* __restrict__ eidx,
                                                       const float* __restrict__ dinv,
                                                       const float* __restrict__ xw1,
                                                       float* __restrict__ h1acc) {
    int t = blockIdx.x * 256 + threadIdx.x;  // 16 lanes per edge
    int e = t >> 4;
    int f = t & 15;
    if (e < NE) {
        int r  = eidx[e];
        int cd = eidx[NE + e];
        float nrm = dinv[r] * dinv[cd];
        atomicAdd(&h1acc[(size_t)cd * HID + f], nrm * xw1[(size_t)r * HID + f]);
    }
}

// ---------------- bias + ReLU (in place) ----------------
__global__ __launch_bounds__(256) void relu_bias_kernel(float* __restrict__ h1,
                                                        const float* __restrict__ b1) {
    int t = blockIdx.x * 256 + threadIdx.x;
    if (t < NN * HID) {
        float v = h1[t] + b1[t & 15];
        h1[t] = v > 0.f ? v : 0.f;
    }
}

// ---------------- GEMM2: xw2 = h1 @ W2 (N padded 40->48, 3 n-tiles) ----------------
// Same dual-tile / dual-chain structure as GEMM1. Per-lane cndmask guards keep
// EXEC all-ones around every v_wmma (ISA requirement); masked stores come after.
__global__ __launch_bounds__(32) void gemm2_wmma_kernel(const float* __restrict__ h1,
                                                        const float* __restrict__ W2,
                                                        const float* __restrict__ dinv,
                                                        float* __restrict__ xw2,
                                                        float* __restrict__ outacc) {
    const int lane = threadIdx.x;
    const int m  = lane & 15;
    const int hi = lane >> 4;
    const int base = blockIdx.x * 32;
    const int n = blockIdx.y * 16 + m;  // output column for this lane (B and C/D)
    const bool nval = (n < NCLS);
    const float* hr0 = h1 + (size_t)(base + m) * HID;
    const float* hr1 = hr0 + (size_t)16 * HID;
    v8f c0 = {0.f, 0.f, 0.f, 0.f, 0.f, 0.f, 0.f, 0.f};
    v8f c1 = {0.f, 0.f, 0.f, 0.f, 0.f, 0.f, 0.f, 0.f};
    for (int k = 0; k < HID; k += 4) {
        const int kk = k + 2 * hi;
        v2f a0 = *(const v2f*)(hr0 + kk);
        v2f a1 = *(const v2f*)(hr1 + kk);
        v2f b;
        b.x = nval ? W2[(size_t)kk * NCLS + n] : 0.f;
        b.y = nval ? W2[(size_t)(kk + 1) * NCLS + n] : 0.f;
        c0 = __builtin_amdgcn_wmma_f32_16x16x4_f32(false, a0, false, b,
                                                   (short)0, c0, false, false);
        c1 = __builtin_amdgcn_wmma_f32_16x16x4_f32(false, a1, false, b,
                                                   (short)0, c1, false, false);
    }
    if (nval) {  // masked stores only, after all WMMAs
#pragma unroll
        for (int t = 0; t < 2; ++t) {
            const v8f& c = t ? c1 : c0;
#pragma unroll
            for (int j = 0; j < 8; ++j) {
                const int r = base + t * 16 + j + 8 * hi;
                const float v  = c[j];
                const float di = dinv[r];
                xw2[(size_t)r * NCLS + n]    = v;            // raw, for edge gather
                outacc[(size_t)r * NCLS + n] = di * di * v;  // self-loop term of layer-2 agg
            }
        }
    }
}

// ---------------- layer-2 edge scatter: one wave per edge, 40 features ----------------
__global__ __launch_bounds__(256) void scatter2_kernel(const int* __restrict__ eidx,
                                                       const float* __restrict__ dinv,
                                                       const float* __restrict__ xw2,
                                                       float* __restrict__ outacc) {
    int w = blockIdx.x * 8 + (threadIdx.x >> 5);
    int lane = threadIdx.x & 31;
    if (w < NE) {
        int r  = eidx[w];
        int cd = eidx[NE + w];
        float nrm = dinv[r] * dinv[cd];
        const float* src = xw2 + (size_t)r * NCLS;
        float* dst = outacc + (size_t)cd * NCLS;
        if (lane < NCLS) atomicAdd(&dst[lane], nrm * src[lane]);
        int f2 = lane + 32;
        if (f2 < NCLS) atomicAdd(&dst[f2], nrm * src[f2]);
    }
}

// ---------------- bias + log_softmax, in place on d_out ----------------
__global__ __launch_bounds__(256) void logsoftmax_kernel(float* __restrict__ out,
                                                         const float* __restrict__ b2) {
    int i = blockIdx.x * 256 + threadIdx.x;
    if (i >= NN) return;
    float* o = out + (size_t)i * NCLS;
    float v[NCLS];
    float mx = -INFINITY;
#pragma unroll
    for (int cc = 0; cc < NCLS; ++cc) {
        v[cc] = o[cc] + b2[cc];
        mx = fmaxf(mx, v[cc]);
    }
    float s = 0.f;
#pragma unroll
    for (int cc = 0; cc < NCLS; ++cc) s += expf(v[cc] - mx);
    float lse = mx + logf(s);
#pragma unroll
    for (int cc = 0; cc < NCLS; ++cc) o[cc] = v[cc] - lse;
}

extern "C" void kernel_launch(void* const* d_in, const int* in_sizes, int n_in,
                              void* d_out, int out_size, void* d_ws, size_t ws_size,
                              hipStream_t stream) {
    const float* x  = (const float*)d_in[0];
    const int* eidx = (const int*)d_in[1];   // [2, NE] flat: rows then cols
    const float* W1 = (const float*)d_in[2];
    const float* b1 = (const float*)d_in[3];
    const float* W2 = (const float*)d_in[4];
    const float* b2 = (const float*)d_in[5];
    float* out = (float*)d_out;

    // workspace carve-up (~29.6 MB), all offsets 8B-aligned for float2 loads
    float* ws   = (float*)d_ws;
    float* deg  = ws;                          // NN
    float* dinv = deg  + NN;                   // NN
    float* xw1  = dinv + NN;                   // NN*HID
    float* h1   = xw1  + (size_t)NN * HID;     // NN*HID
    float* xw2  = h1   + (size_t)NN * HID;     // NN*NCLS

    deg_init_kernel <<<(NN + 255) / 256, 256, 0, stream>>>(deg);
    deg_count_kernel<<<(NE + 255) / 256, 256, 0, stream>>>(eidx + NE, deg);
    dinv_kernel     <<<(NN + 255) / 256, 256, 0, stream>>>(deg, dinv);

    gemm1_wmma_kernel<<<NN / 32, 32, 0, stream>>>(x, W1, dinv, xw1, h1);
    scatter1_kernel  <<<NE * 16 / 256, 256, 0, stream>>>(eidx, dinv, xw1, h1);
    relu_bias_kernel <<<(NN * HID + 255) / 256, 256, 0, stream>>>(h1, b1);

    dim3 g2(NN / 32, (NCLS + 15) / 16);  // 3125 x 3
    gemm2_wmma_kernel<<<g2, 32, 0, stream>>>(h1, W2, dinv, xw2, out);
    scatter2_kernel  <<<(NE + 7) / 8, 256, 0, stream>>>(eidx, dinv, xw2, out);
    logsoftmax_kernel<<<(NN + 255) / 256, 256, 0, stream>>>(out, b2);
}